// KPConvBlock_87239375717066
// MI455X (gfx1250) — compile-verified
//
#include <hip/hip_runtime.h>
#include <hip/hip_bf16.h>

// ---- problem constants (match reference setup_inputs) ----
#define BATCH   4
#define NQ      8192
#define NS      8192
#define HNB     32      // neighbors per query
#define NKP     15      // kernel points (padded to 16 in the WMMA M dim)
#define CIN     64
#define COUT    128
#define TILE_Q  16      // queries per wave
#define WAVES   4       // waves per workgroup
#define TILES_PER_B (NQ / TILE_Q)       // 512
#define NTILES  (BATCH * TILES_PER_B)   // 2048
#define WGT_QSTRIDE 1028                // 16*64 + 4 skew -> bank rotation for stage-2 A reads

typedef float v2f __attribute__((ext_vector_type(2)));
typedef float v8f __attribute__((ext_vector_type(8)));

// ============================================================================
// Kernel 1: fused KPConv  (influence -> weighted -> per-kernel linear + bias)
//   grid = 512 blocks x 128 threads; each wave owns one 16-query tile.
// ============================================================================
__global__ __launch_bounds__(128, 1)
void kpconv_fused(const float* __restrict__ query,
                  const float* __restrict__ support,
                  const float* __restrict__ features,
                  const int*   __restrict__ neighbors,
                  const float* __restrict__ kpts,
                  const float* __restrict__ weights,
                  const float* __restrict__ bias,
                  float*       __restrict__ out)
{
    __shared__ float kp_lds[48];                       // 16 kernel points x 3 (15 real + pad)
    __shared__ float infl_lds[WAVES][HNB * 16];        // [h][kp] per wave
    __shared__ int   nb_lds[WAVES][HNB];               // neighbor ids, current query
    __shared__ float wgt_lds[WAVES][TILE_Q * WGT_QSTRIDE]; // weighted (16q x 16kp x 64c, skewed)

    const int tid  = threadIdx.x;
    const int lane = tid & 31;
    const int wv   = tid >> 5;
    const int hi   = lane >> 4;     // 0: lanes 0-15, 1: lanes 16-31
    const int ln   = lane & 15;

    if (tid < 45) kp_lds[tid] = kpts[tid];
    else if (tid < 48) kp_lds[tid] = 0.0f;
    __syncthreads();

    const int tile = blockIdx.x * WAVES + wv;
    const int b    = tile / TILES_PER_B;
    const int q0   = (tile % TILES_PER_B) * TILE_Q;

    float* wgt  = wgt_lds[wv];
    float* infl = infl_lds[wv];
    int*   nbw  = nb_lds[wv];

    const float* featB = features + (size_t)b * NS * CIN;

    // ---------------- Stage 1: weighted[q][kp][c] via WMMA ------------------
#pragma unroll 1
    for (int m = 0; m < TILE_Q; ++m) {
        const int q = q0 + m;
        // lane == neighbor index h
        const int nidx = neighbors[(size_t)(b * NQ + q) * HNB + lane];
        nbw[lane] = nidx;

        const float qx = query[(size_t)(b * NQ + q) * 3 + 0];
        const float qy = query[(size_t)(b * NQ + q) * 3 + 1];
        const float qz = query[(size_t)(b * NQ + q) * 3 + 2];
        const float* sp = support + (size_t)(b * NS + nidx) * 3;
        const float rx = sp[0] - qx;
        const float ry = sp[1] - qy;
        const float rz = sp[2] - qz;

#pragma unroll
        for (int k = 0; k < NKP; ++k) {
            const float dx = rx - kp_lds[k * 3 + 0];
            const float dy = ry - kp_lds[k * 3 + 1];
            const float dz = rz - kp_lds[k * 3 + 2];
            const float d  = __builtin_sqrtf(dx * dx + dy * dy + dz * dz);
            infl[lane * 16 + k] = fmaxf(0.0f, 1.0f - d);   // SIGMA = 1
        }
        infl[lane * 16 + 15] = 0.0f;                       // zero-pad kp row 15

        // wave-private LDS producer->consumer: wait for DS writes to land
        asm volatile("s_wait_dscnt 0x0" ::: "memory");

        v8f acc[4] = {};   // D = (16 kp x 64 c) in 4 column chunks
#pragma unroll
        for (int s = 0; s < 8; ++s) {                      // K = 32 neighbors, 4 at a time
            const int h = 4 * s + 2 * hi;
            v2f a;                                          // A: infl^T (16kp x 4h)
            a.x = infl[(h + 0) * 16 + ln];
            a.y = infl[(h + 1) * 16 + ln];
            const int i0 = nbw[h + 0];
            const int i1 = nbw[h + 1];
            const float* f0 = featB + (size_t)i0 * CIN;
            const float* f1 = featB + (size_t)i1 * CIN;
#pragma unroll
            for (int n = 0; n < 4; ++n) {                  // 64 cols in 16-chunks
                v2f bf;                                     // B: feats (4h x 16c)
                bf.x = f0[n * 16 + ln];
                bf.y = f1[n * 16 + ln];
                acc[n] = __builtin_amdgcn_wmma_f32_16x16x4_f32(
                    false, a, false, bf, (short)0, acc[n], false, false);
            }
        }
        // D layout: VGPR r -> kp row (r + 8*hi), col = 16n + ln
#pragma unroll
        for (int n = 0; n < 4; ++n) {
#pragma unroll
            for (int r = 0; r < 8; ++r) {
                wgt[m * WGT_QSTRIDE + (r + 8 * hi) * CIN + n * 16 + ln] = acc[n][r];
            }
        }
    }

    asm volatile("s_wait_dscnt 0x0" ::: "memory");

    // ---------------- Stage 2: out(16q x 128) = weighted x W ----------------
    v8f o[8] = {};
#pragma unroll 1
    for (int kp = 0; kp < NKP; ++kp) {
        const float* wk = weights + (size_t)kp * CIN * COUT;
#pragma unroll 1
        for (int cc = 0; cc < 16; ++cc) {                  // Cin in 4-wide K steps
            const int c = cc * 4 + 2 * hi;
            v2f a;                                          // A: weighted (16q x 4c)
            a.x = wgt[ln * WGT_QSTRIDE + kp * CIN + c + 0];
            a.y = wgt[ln * WGT_QSTRIDE + kp * CIN + c + 1];
#pragma unroll
            for (int dc = 0; dc < 8; ++dc) {               // Cout in 16-chunks
                v2f bf;                                     // B: W (4c x 16d)
                bf.x = wk[(c + 0) * COUT + dc * 16 + ln];
                bf.y = wk[(c + 1) * COUT + dc * 16 + ln];
                o[dc] = __builtin_amdgcn_wmma_f32_16x16x4_f32(
                    false, a, false, bf, (short)0, o[dc], false, false);
            }
        }
    }

    // ---------------- epilogue: + bias, store pre-BN output ----------------
    float* ob = out + (size_t)(b * NQ + q0) * COUT;
#pragma unroll
    for (int dc = 0; dc < 8; ++dc) {
        const float bb = bias[dc * 16 + ln];
#pragma unroll
        for (int r = 0; r < 8; ++r) {
            ob[(size_t)(r + 8 * hi) * COUT + dc * 16 + ln] = o[dc][r] + bb;
        }
    }
}

// ============================================================================
// Kernel 2: per-channel batch stats (training-mode mean / rstd over B*Nq)
//   grid = 128 blocks (one per channel) x 256 threads
// ============================================================================
__global__ __launch_bounds__(256)
void bn_stats(const float* __restrict__ out, float* __restrict__ stats)
{
    __shared__ float ssum[256];
    __shared__ float ssq[256];
    const int c = blockIdx.x;
    const int t = threadIdx.x;
    float s = 0.0f, s2 = 0.0f;
    for (int i = t; i < BATCH * NQ; i += 256) {
        const float v = out[(size_t)i * COUT + c];
        s  += v;
        s2 += v * v;
    }
    ssum[t] = s;
    ssq[t]  = s2;
    __syncthreads();
    for (int off = 128; off > 0; off >>= 1) {
        if (t < off) {
            ssum[t] += ssum[t + off];
            ssq[t]  += ssq[t + off];
        }
        __syncthreads();
    }
    if (t == 0) {
        const float inv_n = 1.0f / (float)(BATCH * NQ);
        const float mean  = ssum[0] * inv_n;
        const float var   = ssq[0] * inv_n - mean * mean;
        stats[c]        = mean;
        stats[COUT + c] = rsqrtf(var + 1e-5f);
    }
}

// ============================================================================
// Kernel 3: normalize + affine + ReLU, in place on d_out
// ============================================================================
__global__ __launch_bounds__(256)
void bn_apply(float* __restrict__ out, const float* __restrict__ stats,
              const float* __restrict__ gamma, const float* __restrict__ beta)
{
    const int i = blockIdx.x * 256 + threadIdx.x;  // total = 4*8192*128 = 4194304
    const int c = i & (COUT - 1);
    const float v = out[i];
    const float r = (v - stats[c]) * stats[COUT + c] * gamma[c] + beta[c];
    out[i] = fmaxf(r, 0.0f);
}

// ============================================================================
extern "C" void kernel_launch(void* const* d_in, const int* in_sizes, int n_in,
                              void* d_out, int out_size, void* d_ws, size_t ws_size,
                              hipStream_t stream)
{
    const float* query     = (const float*)d_in[0];
    const float* support   = (const float*)d_in[1];
    const float* features  = (const float*)d_in[2];
    const int*   neighbors = (const int*)  d_in[3];
    const float* kpts      = (const float*)d_in[4];
    const float* weights   = (const float*)d_in[5];
    const float* bias      = (const float*)d_in[6];
    const float* gamma     = (const float*)d_in[7];
    const float* beta      = (const float*)d_in[8];
    float* out   = (float*)d_out;
    float* stats = (float*)d_ws;   // 256 floats: mean[128], rstd[128]

    hipLaunchKernelGGL(kpconv_fused, dim3(NTILES / WAVES), dim3(WAVES * 32), 0, stream,
                       query, support, features, neighbors, kpts, weights, bias, out);
    hipLaunchKernelGGL(bn_stats, dim3(COUT), dim3(256), 0, stream, out, stats);
    hipLaunchKernelGGL(bn_apply, dim3((BATCH * NQ * COUT) / 256), dim3(256), 0, stream,
                       out, stats, gamma, beta);
}